// SelfAttention_43284680409543
// MI455X (gfx1250) — compile-verified
//
#include <hip/hip_runtime.h>

typedef __attribute__((ext_vector_type(2))) float v2f;
typedef __attribute__((ext_vector_type(8))) float v8f;

#define N_TOK 4096   // H*W
#define NB    8      // batch

// ---------------- Kernel 1: per-batch min/max of x -> ws[2b], ws[2b+1] ----
__global__ void mm_minmax_kernel(const float* __restrict__ x,
                                 float* __restrict__ ws) {
  __shared__ float smin[8], smax[8];           // 256 threads = 8 wave32
  const int b = blockIdx.x;
  const float* xb = x + b * N_TOK;
  float vmin = 3.4e38f, vmax = -3.4e38f;
  for (int i = threadIdx.x; i < N_TOK; i += blockDim.x) {
    float v = xb[i];
    vmin = fminf(vmin, v);
    vmax = fmaxf(vmax, v);
  }
  // wave32 butterfly
  #pragma unroll
  for (int off = 16; off > 0; off >>= 1) {
    vmin = fminf(vmin, __shfl_xor(vmin, off, 32));
    vmax = fmaxf(vmax, __shfl_xor(vmax, off, 32));
  }
  const int wave = threadIdx.x >> 5;
  if ((threadIdx.x & 31) == 0) { smin[wave] = vmin; smax[wave] = vmax; }
  __syncthreads();
  if (threadIdx.x == 0) {
    float mn = smin[0], mx = smax[0];
    #pragma unroll
    for (int w = 1; w < 8; ++w) { mn = fminf(mn, smin[w]); mx = fmaxf(mx, smax[w]); }
    ws[2 * b]     = mn;
    ws[2 * b + 1] = mx;
  }
}

// ---------------- Kernel 2: rank-1 attention, one 16-row tile per wave ----
// scores tile (16x16) = q_tile (outer) k_tile via V_WMMA_F32_16X16X4_F32:
//   A (16x4): lanes 0-15, VGPR0 = q[m] at K=0; all other A slots zero.
//   B (4x16): lanes 0-15, VGPR0 = k[n] at K=0; all other B slots zero.
//   D (16x16): VGPR r holds row (r + 8*(lane>=16)), column = lane&15.
__global__ void rank1_attn_kernel(const float* __restrict__ x,
                                  const float* __restrict__ wq_p,
                                  const float* __restrict__ bq_p,
                                  const float* __restrict__ wk_p,
                                  const float* __restrict__ bk_p,
                                  const float* __restrict__ wv_p,
                                  const float* __restrict__ bv_p,
                                  const float* __restrict__ gamma_p,
                                  const float* __restrict__ ws,
                                  float* __restrict__ out) {
  const float wq = wq_p[0], bq = bq_p[0];
  const float wk = wk_p[0], bk = bk_p[0];
  const float wv = wv_p[0], bv = bv_p[0];
  const float gamma = gamma_p[0];

  const int lane = threadIdx.x & 31;
  const int wave = threadIdx.x >> 5;
  const int tile = blockIdx.x * (blockDim.x >> 5) + wave;   // wave-uniform
  const int tiles_per_batch = N_TOK / 16;                   // 256
  const int b    = tile / tiles_per_batch;
  const int row0 = (tile % tiles_per_batch) * 16;
  const float* xb = x + b * N_TOK;

  // k range for closed-form row maxima
  const float xmin = ws[2 * b], xmax = ws[2 * b + 1];
  float kmin = wk * xmin + bk;
  float kmax = wk * xmax + bk;
  if (wk < 0.0f) { float t = kmin; kmin = kmax; kmax = t; }

  // A operand: K=0 column carries the 16 q values (lanes 0-15 only)
  const float q_a = wq * xb[row0 + (lane & 15)] + bq;
  v2f amat;
  amat.x = (lane < 16) ? q_a : 0.0f;   // lanes16-31 are K=2 -> zero
  amat.y = 0.0f;                       // K=1 / K=3 -> zero

  // This lane's 8 rows: m = r + half, with exact softmax max per row
  const int half = (lane >> 4) << 3;   // 0 or 8
  float mrow[8];
  #pragma unroll
  for (int r = 0; r < 8; ++r) {
    float q = wq * xb[row0 + half + r] + bq;
    mrow[r] = (q > 0.0f) ? q * kmax : q * kmin;
  }

  float num[8], den[8];
  #pragma unroll
  for (int r = 0; r < 8; ++r) { num[r] = 0.0f; den[r] = 0.0f; }

  for (int t = 0; t < N_TOK; t += 16) {
    const float xc = xb[t + (lane & 15)];       // column n = lane&15
    const float kc = wk * xc + bk;
    const float vc = wv * xc + bv;

    v2f bmat;
    bmat.x = (lane < 16) ? kc : 0.0f;           // B row K=0 = k tile
    bmat.y = 0.0f;
    v8f c = {};
    // scores tile: D = A x B + 0  (outer product q ⊗ k)
    v8f d = __builtin_amdgcn_wmma_f32_16x16x4_f32(
        /*neg_a=*/false, amat, /*neg_b=*/false, bmat,
        /*c_mod=*/(short)0, c, /*reuse_a=*/false, /*reuse_b=*/false);

    #pragma unroll
    for (int r = 0; r < 8; ++r) {
      const float p = __expf(d[r] - mrow[r]);   // v_exp_f32 path
      num[r] += p * vc;
      den[r] += p;
    }
  }

  // Reduce this lane's column-partials across the 16 lanes of each half.
  #pragma unroll
  for (int off = 1; off < 16; off <<= 1) {
    #pragma unroll
    for (int r = 0; r < 8; ++r) {
      num[r] += __shfl_xor(num[r], off, 32);
      den[r] += __shfl_xor(den[r], off, 32);
    }
  }

  if ((lane & 15) == 0) {
    #pragma unroll
    for (int r = 0; r < 8; ++r) {
      const int row = row0 + half + r;
      const float o = num[r] / den[r];
      out[b * N_TOK + row] = gamma * o + xb[row];
    }
  }
}

// ---------------------------------------------------------------------------
extern "C" void kernel_launch(void* const* d_in, const int* in_sizes, int n_in,
                              void* d_out, int out_size, void* d_ws, size_t ws_size,
                              hipStream_t stream) {
  (void)in_sizes; (void)n_in; (void)out_size; (void)ws_size;
  const float* x     = (const float*)d_in[0];
  const float* wq    = (const float*)d_in[1];
  const float* bq    = (const float*)d_in[2];
  const float* wk    = (const float*)d_in[3];
  const float* bk    = (const float*)d_in[4];
  const float* wv    = (const float*)d_in[5];
  const float* bv    = (const float*)d_in[6];
  const float* gamma = (const float*)d_in[7];
  float* ws  = (float*)d_ws;
  float* out = (float*)d_out;

  mm_minmax_kernel<<<NB, 256, 0, stream>>>(x, ws);

  // 2048 row tiles total, 8 waves (tiles) per 256-thread block -> 256 blocks
  const int total_tiles = NB * (N_TOK / 16);
  rank1_attn_kernel<<<total_tiles / 8, 256, 0, stream>>>(
      x, wq, bq, wk, bk, wv, bv, gamma, ws, out);
}